// GC_FNO3DLoss_53283364274443
// MI455X (gfx1250) — compile-verified
//
#include <hip/hip_runtime.h>
#include <hip/hip_bf16.h>

// ---- problem constants (match reference) ----
#define Bn 8
#define Dn 64
#define Hn 128
#define Wn 64
constexpr int SP    = Dn * Hn * Wn;        // 524288 spatial elems per channel (2^19)
constexpr int SP4   = SP / 4;              // 131072 float4 chunks per channel (2^17)
constexpr int NVEC  = Bn * SP4;            // 1,048,576 float4 chunks total
constexpr int NB    = 1024;                // blocks (multiple of 64 for wmma reduce)
constexpr int NT    = 256;                 // 8 wave32 per block

constexpr float DXf = 2.5f / 63.0f;
constexpr float DYf = 1.0f / 127.0f;
constexpr float DZf = 0.8f / 63.0f;
constexpr float I2DX = 1.0f / (2.0f * DXf);
constexpr float I2DY = 1.0f / (2.0f * DYf);
constexpr float I2DZ = 1.0f / (2.0f * DZf);
constexpr float IDX2 = 1.0f / (DXf * DXf);
constexpr float IDY2 = 1.0f / (DYf * DYf);
constexpr float IDZ2 = 1.0f / (DZf * DZf);
constexpr float INV_RE = 1.0f / 1000000.0f;

typedef __attribute__((ext_vector_type(2))) float v2f;
typedef __attribute__((ext_vector_type(4))) float v4f;
typedef __attribute__((ext_vector_type(8))) float v8f;

// components: 0=field 1=cont 2=mom 3=noslip 4=inlet 5=outlet
__global__ __launch_bounds__(NT) void pinn_loss_main(
    const float* __restrict__ field, const float* __restrict__ gt,
    const float* __restrict__ sdf, float* __restrict__ partials)
{
    float accF = 0.f, accC = 0.f, accM = 0.f, accN = 0.f, accI = 0.f, accO = 0.f;
    const int stride = gridDim.x * blockDim.x;   // 262,144 threads
    for (int cidx = blockIdx.x * blockDim.x + threadIdx.x; cidx < NVEC; cidx += stride) {
        const int spv = cidx & (SP4 - 1);        // chunk index within one channel volume
        const int b   = cidx >> 17;              // SP4 == 2^17
        const int xc  = spv & 15;                // 16 float4 chunks per row
        const int y   = (spv >> 4) & (Hn - 1);
        const int z   = spv >> 11;
        const int sp  = spv * 4;                 // scalar element index (x = sp&63)

        const float* fb  = field + (size_t)b * 4 * SP;
        const float* gb  = gt    + (size_t)b * 4 * SP;
        const v4f*   fb4 = (const v4f*)fb;
        const v4f*   gb4 = (const v4f*)gb;
        const v4f*   sd4 = (const v4f*)(sdf + (size_t)b * SP);

        // prefetch next grid-stride chunk's u-line (gfx1250 global_prefetch_b8)
        {
            int j = cidx + stride;
            if (j < NVEC) {
                int spv2 = j & (SP4 - 1), b2 = j >> 17;
                __builtin_prefetch(field + (size_t)b2 * 4 * SP + spv2 * 4, 0, 0);
            }
        }

        const v4f sd = sd4[spv];
        const v4f u  = fb4[spv];
        const v4f v  = fb4[SP4 + spv];
        const v4f w  = fb4[2 * SP4 + spv];
        const v4f p  = fb4[3 * SP4 + spv];

        // field MSE (4 channels x 4 points)
        {
            v4f du = u - gb4[spv];
            v4f dv = v - gb4[SP4 + spv];
            v4f dw = w - gb4[2 * SP4 + spv];
            v4f dp = p - gb4[3 * SP4 + spv];
            v4f s  = du * du + dv * dv + dw * dw + dp * dp;
            accF += s.x + s.y + s.z + s.w;
        }
        // noslip (solid mask) per point
        #pragma unroll
        for (int i = 0; i < 4; ++i) {
            float solid = (sd[i] <= 0.0f) ? 1.0f : 0.0f;
            accN += (u[i] * u[i] + v[i] * v[i] + w[i] * w[i]) * solid;
        }
        // inlet x==0 lives in chunk 0 element 0
        if (xc == 0) {
            float t = u.x - 1.0f;
            accI += t * t + v.x * v.x + w.x * w.x;
        }
        // outlet x==63 and x==62 both live in chunk 15 (elements 3, 2)
        if (xc == 15) {
            float a0 = u.w - u.z, a1 = v.w - v.z, a2 = w.w - w.z;
            accO += a0 * a0 + a1 * a1 + a2 * a2;
        }
        // interior: continuity + momentum (y,z uniform per chunk; x masked per element)
        if (y >= 1 && y <= Hn - 2 && z >= 1 && z <= Dn - 2) {
            const int OY = 16, OZ = 2048;        // in float4 units
            const v4f uyp = fb4[spv + OY],           uym = fb4[spv - OY];
            const v4f uzp = fb4[spv + OZ],           uzm = fb4[spv - OZ];
            const v4f vyp = fb4[SP4 + spv + OY],     vym = fb4[SP4 + spv - OY];
            const v4f vzp = fb4[SP4 + spv + OZ],     vzm = fb4[SP4 + spv - OZ];
            const v4f wyp = fb4[2*SP4 + spv + OY],   wym = fb4[2*SP4 + spv - OY];
            const v4f wzp = fb4[2*SP4 + spv + OZ],   wzm = fb4[2*SP4 + spv - OZ];
            const v4f pyp = fb4[3*SP4 + spv + OY],   pym = fb4[3*SP4 + spv - OY];
            const v4f pzp = fb4[3*SP4 + spv + OZ],   pzm = fb4[3*SP4 + spv - OZ];

            // x-edge scalars (guarded: only needed when the edge element is interior)
            float uL = 0.f, vL = 0.f, wL = 0.f, pL = 0.f;
            float uR = 0.f, vR = 0.f, wR = 0.f, pR = 0.f;
            if (xc > 0)  { uL = fb[sp - 1]; vL = fb[SP + sp - 1]; wL = fb[2*SP + sp - 1]; pL = fb[3*SP + sp - 1]; }
            if (xc < 15) { uR = fb[sp + 4]; vR = fb[SP + sp + 4]; wR = fb[2*SP + sp + 4]; pR = fb[3*SP + sp + 4]; }

            const float ue[6] = { uL, u.x, u.y, u.z, u.w, uR };
            const float ve[6] = { vL, v.x, v.y, v.z, v.w, vR };
            const float we[6] = { wL, w.x, w.y, w.z, w.w, wR };
            const float pe[6] = { pL, p.x, p.y, p.z, p.w, pR };

            #pragma unroll
            for (int i = 0; i < 4; ++i) {
                const int xi = 4 * xc + i;
                const float inter = (xi >= 1 && xi <= Wn - 2) ? 1.0f : 0.0f;
                const float fluid = (sd[i] > 0.0f) ? inter : 0.0f;

                const float uc = ue[i + 1], vc = ve[i + 1], wc = we[i + 1];
                const float du_dx = (ue[i + 2] - ue[i]) * I2DX;
                const float dv_dx = (ve[i + 2] - ve[i]) * I2DX;
                const float dw_dx = (we[i + 2] - we[i]) * I2DX;
                const float dp_dx = (pe[i + 2] - pe[i]) * I2DX;
                const float du_dy = (uyp[i] - uym[i]) * I2DY;
                const float dv_dy = (vyp[i] - vym[i]) * I2DY;
                const float dw_dy = (wyp[i] - wym[i]) * I2DY;
                const float dp_dy = (pyp[i] - pym[i]) * I2DY;
                const float du_dz = (uzp[i] - uzm[i]) * I2DZ;
                const float dv_dz = (vzp[i] - vzm[i]) * I2DZ;
                const float dw_dz = (wzp[i] - wzm[i]) * I2DZ;
                const float dp_dz = (pzp[i] - pzm[i]) * I2DZ;

                const float lap_u = (ue[i+2] - 2.f*uc + ue[i])*IDX2 + (uyp[i] - 2.f*uc + uym[i])*IDY2 + (uzp[i] - 2.f*uc + uzm[i])*IDZ2;
                const float lap_v = (ve[i+2] - 2.f*vc + ve[i])*IDX2 + (vyp[i] - 2.f*vc + vym[i])*IDY2 + (vzp[i] - 2.f*vc + vzm[i])*IDZ2;
                const float lap_w = (we[i+2] - 2.f*wc + we[i])*IDX2 + (wyp[i] - 2.f*wc + wym[i])*IDY2 + (wzp[i] - 2.f*wc + wzm[i])*IDZ2;

                const float div = du_dx + dv_dy + dw_dz;
                const float dm  = div * fluid;
                accC += dm * dm;

                const float rx = uc*du_dx + vc*du_dy + wc*du_dz + dp_dx - INV_RE*lap_u;
                const float ry = uc*dv_dx + vc*dv_dy + wc*dv_dz + dp_dy - INV_RE*lap_v;
                const float rz = uc*dw_dx + vc*dw_dy + wc*dw_dz + dp_dz - INV_RE*lap_w;
                accM += fluid * (rx * rx + ry * ry + rz * rz);
            }
        }
    }

    // wave32 butterfly + cross-wave LDS reduction (deterministic, no atomics)
    float a6[6] = { accF, accC, accM, accN, accI, accO };
    #pragma unroll
    for (int c = 0; c < 6; ++c) {
        float s = a6[c];
        #pragma unroll
        for (int off = 16; off >= 1; off >>= 1) s += __shfl_xor(s, off, 32);
        a6[c] = s;
    }
    __shared__ float sm[NT / 32][6];
    const int wid = threadIdx.x >> 5, ln = threadIdx.x & 31;
    if (ln == 0) {
        #pragma unroll
        for (int c = 0; c < 6; ++c) sm[wid][c] = a6[c];
    }
    __syncthreads();
    if (threadIdx.x == 0) {
        #pragma unroll
        for (int c = 0; c < 6; ++c) {
            float t = 0.f;
            #pragma unroll
            for (int w8 = 0; w8 < NT / 32; ++w8) t += sm[w8][c];
            partials[c * NB + blockIdx.x] = t;
        }
    }
}

// One wave32. Reduces 6×NB partials with V_WMMA_F32_16X16X4_F32 (A=partials tile,
// B=ones -> row sums), adds the ct MSE, and emits the weighted scalar.
__global__ __launch_bounds__(32) void pinn_loss_final(
    const float* __restrict__ partials, const float* __restrict__ ct,
    const float* __restrict__ gt_ct, float* __restrict__ out)
{
    const int lane = threadIdx.x;
    const int m  = lane & 15;                 // A-matrix row held by this lane
    const int k0 = (lane < 16) ? 0 : 2;       // K=0,1 in lanes 0-15; K=2,3 in lanes 16-31
    v2f ones; ones.x = 1.0f; ones.y = 1.0f;

    float totals[6];
    #pragma unroll
    for (int c = 0; c < 6; ++c) {
        const float* pc = partials + c * NB;
        v8f acc = {};
        for (int chunk = 0; chunk < NB; chunk += 64) {
            v2f a;                             // chunk viewed as row-major 16x4
            a.x = pc[chunk + m * 4 + k0];
            a.y = pc[chunk + m * 4 + k0 + 1];
            // D[r][n] = sum_k A[r][k]  (EXEC all-ones, uniform flow)
            acc = __builtin_amdgcn_wmma_f32_16x16x4_f32(
                false, a, false, ones, (short)0, acc, false, false);
        }
        float s = acc[0] + acc[1] + acc[2] + acc[3] + acc[4] + acc[5] + acc[6] + acc[7];
        #pragma unroll
        for (int off = 16; off >= 1; off >>= 1) s += __shfl_xor(s, off, 32);
        totals[c] = s * (1.0f / 16.0f);        // full-lane sum counts every D element 16x
    }

    // ct MSE over 8 elements (branchless lane masking, EXEC stays full until the end)
    {
        int i = lane & 7;
        float d = ct[i] - gt_ct[i];
        float vv = (lane < 8) ? d * d : 0.0f;
        #pragma unroll
        for (int off = 16; off >= 1; off >>= 1) vv += __shfl_xor(vv, off, 32);
        if (lane == 0) {
            const float loss_ct = vv * (1.0f / 8.0f);
            float total =
                  totals[0] * (1.0f / 16777216.0f)            // L_FIELD * mean
                + 10.0f * loss_ct                              // L_CT
                + totals[1] * (1.0f / 3874752.0f)              // L_CONT (8*62*126*62)
                + totals[2] * (1.0f / 3874752.0f)              // L_MOM
                + 10.0f * totals[3] * (1.0f / 4194304.0f)      // L_NOSLIP
                + 5.0f  * totals[4] * (1.0f / 65536.0f)        // L_INLET (8*64*128)
                + totals[5] * (1.0f / 65536.0f);               // L_OUTLET
            out[0] = total;
        }
    }
}

extern "C" void kernel_launch(void* const* d_in, const int* in_sizes, int n_in,
                              void* d_out, int out_size, void* d_ws, size_t ws_size,
                              hipStream_t stream) {
    (void)in_sizes; (void)n_in; (void)out_size; (void)ws_size;
    const float* field    = (const float*)d_in[0];
    const float* ct       = (const float*)d_in[1];
    const float* gt_field = (const float*)d_in[2];
    const float* gt_ct    = (const float*)d_in[3];
    const float* sdf      = (const float*)d_in[4];
    float* partials = (float*)d_ws;            // 6 * NB * 4 B = 24 KiB, fully rewritten each call

    pinn_loss_main<<<NB, NT, 0, stream>>>(field, gt_field, sdf, partials);
    pinn_loss_final<<<1, 32, 0, stream>>>(partials, ct, gt_ct, (float*)d_out);
}